// GNN_60430189854726
// MI455X (gfx1250) — compile-verified
//
#include <hip/hip_runtime.h>
#include <hip/hip_bf16.h>

#define N_NODES 50000
#define DIM 128
#define ROW_TILES (N_NODES / 16)   // 3125, exact

typedef __attribute__((ext_vector_type(2))) float v2f;
typedef __attribute__((ext_vector_type(8))) float v8f;

// ---------------------------------------------------------------------------
// Zero-fill workspace regions (graph-capture safe; no memset)
// ---------------------------------------------------------------------------
__global__ __launch_bounds__(256) void sage_fill0(float* __restrict__ p, size_t n) {
    size_t i = (size_t)blockIdx.x * 256 + threadIdx.x;
    if (i < n) p[i] = 0.0f;
}

// ---------------------------------------------------------------------------
// Degree count: one thread per edge, f32 atomic add into count[dst]
// ---------------------------------------------------------------------------
__global__ __launch_bounds__(256) void sage_count(const int* __restrict__ dst,
                                                  float* __restrict__ cnt, int E) {
    int e = blockIdx.x * 256 + threadIdx.x;
    if (e < E) {
        __hip_atomic_fetch_add(&cnt[dst[e]], 1.0f,
                               __ATOMIC_RELAXED, __HIP_MEMORY_SCOPE_AGENT);
    }
}

// count -> 1/max(count,1)   (isolated nodes -> agg stays 0, matching reference)
__global__ __launch_bounds__(256) void sage_recip(float* __restrict__ c, int n) {
    int i = blockIdx.x * 256 + threadIdx.x;
    if (i < n) c[i] = 1.0f / fmaxf(c[i], 1.0f);
}

// ---------------------------------------------------------------------------
// Scatter-add: one wave32 per edge; lane loads float4 (512B row = 32 lanes x 16B),
// then 4 relaxed agent-scope f32 atomics into summed[dst] (L2-resident).
// ---------------------------------------------------------------------------
__global__ __launch_bounds__(256) void sage_scatter(const float* __restrict__ xin,
                                                    const int* __restrict__ src,
                                                    const int* __restrict__ dst,
                                                    float* __restrict__ summed, int E) {
    int gw   = (blockIdx.x * 256 + threadIdx.x) >> 5;   // global wave id = edge id
    int lane = threadIdx.x & 31;
    if (gw >= E) return;
    int s = src[gw];
    int d = dst[gw];
    const float4* xr = (const float4*)(xin + (size_t)s * DIM);
    float4 v = xr[lane];                                // global_load_b128
    float* o = summed + (size_t)d * DIM + lane * 4;
    __hip_atomic_fetch_add(&o[0], v.x, __ATOMIC_RELAXED, __HIP_MEMORY_SCOPE_AGENT);
    __hip_atomic_fetch_add(&o[1], v.y, __ATOMIC_RELAXED, __HIP_MEMORY_SCOPE_AGENT);
    __hip_atomic_fetch_add(&o[2], v.z, __ATOMIC_RELAXED, __HIP_MEMORY_SCOPE_AGENT);
    __hip_atomic_fetch_add(&o[3], v.w, __ATOMIC_RELAXED, __HIP_MEMORY_SCOPE_AGENT);
}

// ---------------------------------------------------------------------------
// Pack W = [Wl ; Wr] (K=256, N=128) transposed and pair-packed along K:
// Wp[p][n] = { W[2p][n], W[2p+1][n] }  so a B fragment for v_wmma_f32_16x16x4_f32
// (lane: N = lane&15, K = {kk,kk+1} | {kk+2,kk+3}) is ONE contiguous float2 load.
// ---------------------------------------------------------------------------
__global__ __launch_bounds__(256) void sage_packw(const float* __restrict__ Wl,
                                                  const float* __restrict__ Wr,
                                                  float2* __restrict__ Wp) {
    int t = blockIdx.x * 256 + threadIdx.x;             // 0 .. 128*128-1
    if (t >= 128 * 128) return;
    int p = t >> 7;                                     // K-pair index 0..127
    int n = t & 127;
    int k0 = 2 * p;
    float a, b;
    if (k0 < DIM) {                                     // Wl part: B[k][n] = Wl[n][k]
        a = Wl[n * DIM + k0];
        b = Wl[n * DIM + k0 + 1];
    } else {                                            // Wr part
        a = Wr[n * DIM + (k0 - DIM)];
        b = Wr[n * DIM + (k0 - DIM) + 1];
    }
    Wp[t] = make_float2(a, b);
}

// ---------------------------------------------------------------------------
// Fused SAGE GEMM: out = [agg | x] @ [Wl;Wr]^T + bl  (K = 256), optional ReLU.
// Block = 256 thr = 8 waves; block computes a 16-row x 128-col tile.
// A panel (16x256 f32) staged in LDS as float2 pairs, padded stride 130 pairs
// (260 f32) -> conflict-free ds_load_b64 fragment reads.
// Wave w computes cols [16w,16w+16) with 64 x v_wmma_f32_16x16x4_f32.
// ---------------------------------------------------------------------------
__global__ __launch_bounds__(256) void sage_gemm(const float* __restrict__ summed,
                                                 const float* __restrict__ inv,
                                                 const float* __restrict__ xin,
                                                 const float2* __restrict__ Wp,
                                                 const float* __restrict__ bias,
                                                 float* __restrict__ out, int relu) {
    __shared__ float2 ldsA[16][130];                    // 16.6 KB

    const int tid = threadIdx.x;
    const int r0  = blockIdx.x * 16;                    // first row of tile

    // Stage A_cat[16][256] as 128 float2 pairs per row (mean-normalized agg | x)
    for (int idx = tid; idx < 16 * 128; idx += 256) {
        int m   = idx >> 7;                             // row in tile
        int p   = idx & 127;                            // K-pair
        int row = r0 + m;
        float2 v;
        if (p < 64) {
            float2 s = ((const float2*)summed)[(size_t)row * 64 + p];
            float iv = inv[row];
            v.x = s.x * iv;
            v.y = s.y * iv;
        } else {
            v = ((const float2*)xin)[(size_t)row * 64 + (p - 64)];
        }
        ldsA[m][p] = v;
    }
    __syncthreads();

    const int lane = tid & 31;
    const int wave = tid >> 5;                          // 0..7 -> column slab
    const int half = lane >> 4;                         // lane 0-15 / 16-31
    const int m16  = lane & 15;
    const int n0   = wave * 16;

    const v2f* wpv = (const v2f*)Wp;

    v8f acc = {};
    #pragma unroll 4
    for (int kp = 0; kp < 128; kp += 2) {               // K advances 4 per wmma
        // A fragment: M = m16, K = {2kp, 2kp+1} (half 0) | {2kp+2, 2kp+3} (half 1)
        v2f a = *(const v2f*)&ldsA[m16][kp + half];     // ds_load_b64
        // B fragment: N = n0+m16, same K pair
        v2f b = wpv[(size_t)(kp + half) * 128 + n0 + m16]; // global_load_b64
        acc = __builtin_amdgcn_wmma_f32_16x16x4_f32(
            /*neg_a=*/false, a, /*neg_b=*/false, b,
            /*c_mod=*/(short)0, acc, /*reuse_a=*/false, /*reuse_b=*/false);
    }

    // D layout: VGPR r -> row = r + 8*half, col = m16 (+ n0)
    const int   ncol = n0 + m16;
    const float bv   = bias[ncol];
    #pragma unroll
    for (int r = 0; r < 8; r++) {
        int   row = r0 + r + 8 * half;
        float val = acc[r] + bv;
        if (relu) val = fmaxf(val, 0.0f);
        out[(size_t)row * DIM + ncol] = val;
    }
}

// ---------------------------------------------------------------------------
// Launch: two SAGE layers, counts computed once.
// ws layout: [0,200KB) counts/inv | [256KB,384KB) packed W |
//            [512KB, +26MB) summed | [+26MB, +52MB) h1
// ---------------------------------------------------------------------------
extern "C" void kernel_launch(void* const* d_in, const int* in_sizes, int n_in,
                              void* d_out, int out_size, void* d_ws, size_t ws_size,
                              hipStream_t stream) {
    const float* x  = (const float*)d_in[0];
    const int*   ei = (const int*)d_in[1];     // [2][E] row-major
    const float* Wl = (const float*)d_in[2];
    const float* bl = (const float*)d_in[3];
    const float* Wr = (const float*)d_in[4];
    const int E = in_sizes[1] / 2;
    const int* src = ei;
    const int* dst = ei + E;

    char*   ws     = (char*)d_ws;
    float*  inv    = (float*)ws;                                    // N f32
    float2* wp     = (float2*)(ws + (256u << 10));                  // 128 KB
    float*  summed = (float*)(ws + (512u << 10));                   // 25.6 MB
    float*  h1     = (float*)(ws + (512u << 10) + (26u << 20));     // 25.6 MB
    float*  y      = (float*)d_out;

    const size_t nfeat = (size_t)N_NODES * DIM;
    const int fillFeatBlocks = (int)((nfeat + 255) / 256);
    const int fillNBlocks    = (N_NODES + 255) / 256;
    const int cntBlocks      = (E + 255) / 256;
    const int scatBlocks     = (E * 32 + 255) / 256;                // 1 wave/edge

    // ---- shared prep: degrees + packed weights -----------------------------
    sage_fill0<<<fillNBlocks, 256, 0, stream>>>(inv, (size_t)N_NODES);
    sage_count<<<cntBlocks, 256, 0, stream>>>(dst, inv, E);
    sage_recip<<<fillNBlocks, 256, 0, stream>>>(inv, N_NODES);
    sage_packw<<<64, 256, 0, stream>>>(Wl, Wr, wp);

    // ---- layer 1: h1 = relu(sage(x)) ---------------------------------------
    sage_fill0<<<fillFeatBlocks, 256, 0, stream>>>(summed, nfeat);
    sage_scatter<<<scatBlocks, 256, 0, stream>>>(x, src, dst, summed, E);
    sage_gemm<<<ROW_TILES, 256, 0, stream>>>(summed, inv, x, wp, bl, h1, 1);

    // ---- layer 2: y = sage(h1) ---------------------------------------------
    sage_fill0<<<fillFeatBlocks, 256, 0, stream>>>(summed, nfeat);
    sage_scatter<<<scatBlocks, 256, 0, stream>>>(h1, src, dst, summed, E);
    sage_gemm<<<ROW_TILES, 256, 0, stream>>>(summed, inv, h1, wp, bl, y, 0);
}